// VisionTransformerForSegmentationMultiScale_28252294873700
// MI455X (gfx1250) — compile-verified
//
#include <hip/hip_runtime.h>
#include <math.h>

#define BB    8
#define IMGSZ 128
#define DIM   128
#define NHEAD 4
#define NLAY  2
#define NCT   256
#define NFT   1024
#define NTOK  1280
#define HD    32

typedef __attribute__((ext_vector_type(2))) float v2f;
typedef __attribute__((ext_vector_type(8))) float v8f;

static __device__ __forceinline__ v8f wmma_f32(v2f a, v2f b, v8f c) {
  // D = A(16x4 f32) * B(4x16 f32) + C(16x16 f32)
  return __builtin_amdgcn_wmma_f32_16x16x4_f32(false, a, false, b, (short)0, c,
                                               false, false);
}

static __device__ __forceinline__ float red_max16(float v) {
  v = fmaxf(v, __shfl_xor(v, 1, 32));
  v = fmaxf(v, __shfl_xor(v, 2, 32));
  v = fmaxf(v, __shfl_xor(v, 4, 32));
  v = fmaxf(v, __shfl_xor(v, 8, 32));
  return v;
}
static __device__ __forceinline__ float red_sum16(float v) {
  v += __shfl_xor(v, 1, 32);
  v += __shfl_xor(v, 2, 32);
  v += __shfl_xor(v, 4, 32);
  v += __shfl_xor(v, 8, 32);
  return v;
}
static __device__ __forceinline__ float red_sum32(float v) {
  v = red_sum16(v);
  v += __shfl_xor(v, 16, 32);
  return v;
}

// ---------------- patch embedding (tiny K: scalar) ----------------
__global__ void coarse_embed_kernel(const float* __restrict__ x,
                                    const float* __restrict__ Wpc,
                                    const float* __restrict__ bpc,
                                    const float* __restrict__ te0,
                                    float* __restrict__ tokens) {
  int idx = blockIdx.x * blockDim.x + threadIdx.x;   // B*256*128 = 2^18
  if (idx >= BB * NCT * DIM) return;
  int d = idx & 127, t = (idx >> 7) & 255, b = idx >> 15;
  int ph = t >> 4, pw = t & 15;
  const float* xb = x + (size_t)b * IMGSZ * IMGSZ + (ph * 8) * IMGSZ + pw * 8;
  float acc = bpc[d] + te0[d];
  for (int i = 0; i < 8; i++)
    for (int j = 0; j < 8; j++)
      acc += xb[i * IMGSZ + j] * Wpc[(i * 8 + j) * DIM + d];
  tokens[((size_t)b * NTOK + t) * DIM + d] = acc;
}

__global__ void fine_embed_kernel(const float* __restrict__ x,
                                  const float* __restrict__ Wpf,
                                  const float* __restrict__ bpf,
                                  float* __restrict__ fineb) {
  int idx = blockIdx.x * blockDim.x + threadIdx.x;   // B*1024*128 = 2^20
  if (idx >= BB * NFT * DIM) return;
  int d = idx & 127, t = (idx >> 7) & 1023, b = idx >> 17;
  int ph = t >> 5, pw = t & 31;
  const float* xb = x + (size_t)b * IMGSZ * IMGSZ + (ph * 4) * IMGSZ + pw * 4;
  float acc = bpf[d];
  for (int i = 0; i < 4; i++)
    for (int j = 0; j < 4; j++)
      acc += xb[i * IMGSZ + j] * Wpf[(i * 4 + j) * DIM + d];
  fineb[((size_t)b * NFT + t) * DIM + d] = acc;
}

// ---------------- sobel -> avgpool(4) -> per-image-mean threshold ----------------
__global__ void edge_mask_kernel(const float* __restrict__ x, int* __restrict__ mask) {
  int b = blockIdx.x;
  int t = threadIdx.x;                 // 1024 cells of 32x32 grid
  int ci = t >> 5, cj = t & 31;
  const float* xb = x + (size_t)b * IMGSZ * IMGSZ;
  float acc = 0.f;
  for (int dy = 0; dy < 4; dy++) {
    for (int dx = 0; dx < 4; dx++) {
      int y = ci * 4 + dy, xx = cj * 4 + dx;
      float v[3][3];
      for (int a = -1; a <= 1; a++)
        for (int c = -1; c <= 1; c++) {
          int yy = y + a, xc = xx + c;
          v[a + 1][c + 1] = (yy >= 0 && yy < IMGSZ && xc >= 0 && xc < IMGSZ)
                                ? xb[yy * IMGSZ + xc] : 0.f;
        }
      float sx = -v[0][0] + v[0][2] - 2.f * v[1][0] + 2.f * v[1][2] - v[2][0] + v[2][2];
      float sy = -v[0][0] - 2.f * v[0][1] - v[0][2] + v[2][0] + 2.f * v[2][1] + v[2][2];
      acc += sqrtf(sx * sx + sy * sy);
    }
  }
  acc *= (1.f / 16.f);
  __shared__ float sm[1024];
  sm[t] = acc;
  __syncthreads();
  for (int off = 512; off > 0; off >>= 1) {
    if (t < off) sm[t] += sm[t + off];
    __syncthreads();
  }
  float mean = sm[0] * (1.f / 1024.f);
  mask[b * NFT + t] = (acc > mean) ? 1 : 0;
}

// ---------------- stable "edge tokens first" ordering via prefix scan ----------------
__global__ void sort_kernel(const int* __restrict__ mask, int* __restrict__ order,
                            int* __restrict__ inv, int* __restrict__ msort) {
  int b = blockIdx.x, i = threadIdx.x;
  __shared__ int s[1024];
  int m = mask[b * NFT + i];
  s[i] = m;
  __syncthreads();
  for (int off = 1; off < 1024; off <<= 1) {
    int add = (i >= off) ? s[i - off] : 0;
    __syncthreads();
    s[i] += add;
    __syncthreads();
  }
  int incl = s[i];
  int total = s[1023];
  int excl = incl - m;
  int pos = m ? excl : (total + (i - excl));
  order[b * NFT + pos] = i;
  inv[b * NFT + i] = pos;
  msort[b * NFT + pos] = m;
}

__global__ void gather_fine_kernel(const float* __restrict__ fineb,
                                   const int* __restrict__ order,
                                   const int* __restrict__ msort,
                                   const float* __restrict__ te1,
                                   float* __restrict__ tokens) {
  int idx = blockIdx.x * blockDim.x + threadIdx.x;   // B*1024*128
  if (idx >= BB * NFT * DIM) return;
  int d = idx & 127, p = (idx >> 7) & 1023, b = idx >> 17;
  int o = order[b * NFT + p];
  float v = msort[b * NFT + p] ? fineb[((size_t)b * NFT + o) * DIM + d] : 0.f;
  tokens[((size_t)b * NTOK + NCT + p) * DIM + d] = v + te1[d];
}

// ---------------- LayerNorm (one wave per 128-wide row) ----------------
__global__ void ln_kernel(const float* __restrict__ in, const float* __restrict__ g,
                          const float* __restrict__ bt, float* __restrict__ out) {
  int wid = (blockIdx.x * blockDim.x + threadIdx.x) >> 5;
  int lane = threadIdx.x & 31;
  if (wid >= BB * NTOK) return;
  const float* xr = in + (size_t)wid * DIM;
  float v0 = xr[lane], v1 = xr[lane + 32], v2 = xr[lane + 64], v3 = xr[lane + 96];
  float s = v0 + v1 + v2 + v3;
  float ss = v0 * v0 + v1 * v1 + v2 * v2 + v3 * v3;
  s = red_sum32(s);
  ss = red_sum32(ss);
  float mean = s * (1.f / 128.f);
  float var = ss * (1.f / 128.f) - mean * mean;
  float r = rsqrtf(var + 1e-5f);
  float* orow = out + (size_t)wid * DIM;
  orow[lane]      = (v0 - mean) * r * g[lane]      + bt[lane];
  orow[lane + 32] = (v1 - mean) * r * g[lane + 32] + bt[lane + 32];
  orow[lane + 64] = (v2 - mean) * r * g[lane + 64] + bt[lane + 64];
  orow[lane + 96] = (v3 - mean) * r * g[lane + 96] + bt[lane + 96];
}

// ---------------- generic f32 WMMA GEMM: C[M,N] = A[M,K] @ B[K,N] ----------------
// One wave computes a 16x64 strip (4 N-tiles) so the A fragment is reused 4x
// per k-step: 4 WMMAs per (2 A + 8 B) loaded floats per lane.
// mode 0: C = r + bias ; mode 1: C = gelu(r + bias) ; mode 2: C += r + bias
// Requires N % 64 == 0, M % 16 == 0, K % 4 == 0.
__global__ void gemm_wmma_kernel(const float* __restrict__ A,
                                 const float* __restrict__ Bm,
                                 const float* __restrict__ bias,
                                 float* __restrict__ C, int N, int K, int mode) {
  int mt = blockIdx.x, nstrip = blockIdx.y;
  int lane = threadIdx.x;
  int l16 = lane & 15, hg = lane >> 4;
  int row = mt * 16 + l16;
  int col0 = nstrip * 64 + l16;
  const float* ar = A + (size_t)row * K;
  v8f c0 = {}, c1 = {}, c2 = {}, c3 = {};
#pragma unroll 4
  for (int k0 = 0; k0 < K; k0 += 4) {
    int k = k0 + 2 * hg;
    v2f a; a.x = ar[k]; a.y = ar[k + 1];
    const float* b0 = Bm + (size_t)k * N + col0;
    const float* b1 = Bm + (size_t)(k + 1) * N + col0;
    v2f bf;
    bf.x = b0[0];  bf.y = b1[0];  c0 = wmma_f32(a, bf, c0);
    bf.x = b0[16]; bf.y = b1[16]; c1 = wmma_f32(a, bf, c1);
    bf.x = b0[32]; bf.y = b1[32]; c2 = wmma_f32(a, bf, c2);
    bf.x = b0[48]; bf.y = b1[48]; c3 = wmma_f32(a, bf, c3);
  }
  v8f acc[4] = {c0, c1, c2, c3};
  for (int g = 0; g < 4; g++) {
    int col = col0 + 16 * g;
    float bv = bias ? bias[col] : 0.f;
    for (int r = 0; r < 8; r++) {
      int orow = mt * 16 + r + 8 * hg;
      size_t idx = (size_t)orow * N + col;
      float val = acc[g][r] + bv;
      if (mode == 1) val = 0.5f * val * (1.f + erff(val * 0.70710678118654752f));
      if (mode == 2) C[idx] += val; else C[idx] = val;
    }
  }
}

// ---------------- attention: per (b, h, 16-row q tile), two-pass softmax ----------------
__global__ void attn_kernel(const float* __restrict__ qkv, float* __restrict__ attnout) {
  int mt = blockIdx.x;            // q tile 0..79
  int h = blockIdx.y;
  int b = blockIdx.z;
  int lane = threadIdx.x;
  int l16 = lane & 15, hg = lane >> 4;
  const float scale = 0.1767766952966369f;   // 32^-0.5
  const float* base = qkv + (size_t)b * NTOK * 384;
  // Q fragments: row = mt*16 + l16, K over hd=32 in 8 steps of 4
  v2f qa[8];
  {
    const float* qp = base + (size_t)(mt * 16 + l16) * 384 + h * HD;
    for (int s = 0; s < 8; s++) {
      int k = 4 * s + 2 * hg;
      v2f t; t.x = qp[k]; t.y = qp[k + 1];
      qa[s] = t;
    }
  }
  float rm[8], rl[8];
  for (int r = 0; r < 8; r++) { rm[r] = -3.0e38f; rl[r] = 0.f; }
  // pass 1: row max + rescaled sum of exp
  for (int nt = 0; nt < NTOK / 16; nt++) {
    const float* kp = base + (size_t)(nt * 16 + l16) * 384 + DIM + h * HD;
    v8f c = {};
    for (int s = 0; s < 8; s++) {
      int k = 4 * s + 2 * hg;
      v2f bf; bf.x = kp[k]; bf.y = kp[k + 1];
      c = wmma_f32(qa[s], bf, c);
    }
    for (int r = 0; r < 8; r++) {
      float v = c[r] * scale;
      float tm = red_max16(v);
      float nm = fmaxf(rm[r], tm);
      float p = expf(v - nm);
      float ts = red_sum16(p);
      rl[r] = rl[r] * expf(rm[r] - nm) + ts;
      rm[r] = nm;
    }
  }
  // pass 2: O = sum_t P_t @ V_t, P in A-layout via LDS transpose
  v8f o0 = {}, o1 = {};
  __shared__ float pt[16][17];
  for (int nt = 0; nt < NTOK / 16; nt++) {
    const float* kp = base + (size_t)(nt * 16 + l16) * 384 + DIM + h * HD;
    v8f c = {};
    for (int s = 0; s < 8; s++) {
      int k = 4 * s + 2 * hg;
      v2f bf; bf.x = kp[k]; bf.y = kp[k + 1];
      c = wmma_f32(qa[s], bf, c);
    }
    __syncthreads();
    for (int r = 0; r < 8; r++)
      pt[r + 8 * hg][l16] = expf(c[r] * scale - rm[r]);
    __syncthreads();
    const float* vbase = base + 2 * DIM + h * HD;
    for (int s = 0; s < 4; s++) {
      int k = 4 * s + 2 * hg;
      v2f pa; pa.x = pt[l16][k]; pa.y = pt[l16][k + 1];
      const float* v0p = vbase + (size_t)(nt * 16 + k) * 384;
      const float* v1p = vbase + (size_t)(nt * 16 + k + 1) * 384;
      v2f vb; vb.x = v0p[l16]; vb.y = v1p[l16];
      o0 = wmma_f32(pa, vb, o0);
      v2f vb2; vb2.x = v0p[16 + l16]; vb2.y = v1p[16 + l16];
      o1 = wmma_f32(pa, vb2, o1);
    }
  }
  float* ob = attnout + (size_t)b * NTOK * DIM + h * HD;
  for (int r = 0; r < 8; r++) {
    int row = mt * 16 + r + 8 * hg;
    float invl = 1.0f / rl[r];
    ob[(size_t)row * DIM + l16] = o0[r] * invl;
    ob[(size_t)row * DIM + 16 + l16] = o1[r] * invl;
  }
}

// ---------------- decoder heads (D=128 dot per token, one wave each) ----------------
__global__ void head_kernel(const float* __restrict__ tokens,
                            const float* __restrict__ dWc, const float* __restrict__ dbc,
                            const float* __restrict__ dWf, const float* __restrict__ dbf,
                            const int* __restrict__ msort,
                            float* __restrict__ cmap, float* __restrict__ feats) {
  int wid = (blockIdx.x * blockDim.x + threadIdx.x) >> 5;
  int lane = threadIdx.x & 31;
  if (wid >= BB * NTOK) return;
  int b = wid / NTOK, t = wid % NTOK;
  const float* row = tokens + (size_t)wid * DIM;
  const float* W = (t < NCT) ? dWc : dWf;
  float s = row[lane] * W[lane] + row[lane + 32] * W[lane + 32] +
            row[lane + 64] * W[lane + 64] + row[lane + 96] * W[lane + 96];
  s = red_sum32(s);
  if (lane == 0) {
    if (t < NCT) cmap[b * NCT + t] = s + dbc[0];
    else {
      int p = t - NCT;
      feats[b * NFT + p] = (s + dbf[0]) * (msort[b * NFT + p] ? 1.f : 0.f);
    }
  }
}

__global__ void scatter_kernel(const float* __restrict__ feats,
                               const int* __restrict__ inv, float* __restrict__ fflat) {
  int idx = blockIdx.x * blockDim.x + threadIdx.x;
  if (idx >= BB * NFT) return;
  int b = idx >> 10;
  fflat[idx] = feats[b * NFT + inv[idx]];
}

// c_up = bilinear(16->32, align_corners); f_up = identity (32->32)
__global__ void fuse_kernel(const float* __restrict__ cmap,
                            const float* __restrict__ fflat, float* __restrict__ fused) {
  int idx = blockIdx.x * blockDim.x + threadIdx.x;
  if (idx >= BB * 1024) return;
  int b = idx >> 10, oy = (idx >> 5) & 31, ox = idx & 31;
  float fy = oy * (15.f / 31.f);
  int y0 = (int)floorf(fy); int y1 = min(y0 + 1, 15); float wy = fy - (float)y0;
  float fx = ox * (15.f / 31.f);
  int x0 = (int)floorf(fx); int x1 = min(x0 + 1, 15); float wx = fx - (float)x0;
  const float* cm = cmap + b * NCT;
  float cv = cm[y0 * 16 + x0] * (1.f - wy) * (1.f - wx) +
             cm[y0 * 16 + x1] * (1.f - wy) * wx +
             cm[y1 * 16 + x0] * wy * (1.f - wx) +
             cm[y1 * 16 + x1] * wy * wx;
  fused[(size_t)b * 2048 + oy * 32 + ox] = cv;
  fused[(size_t)b * 2048 + 1024 + oy * 32 + ox] = fflat[idx];
}

__global__ void conv1_kernel(const float* __restrict__ fused,
                             const float* __restrict__ fW1, const float* __restrict__ fb1,
                             float* __restrict__ hbuf) {
  int idx = blockIdx.x * blockDim.x + threadIdx.x;   // B*2*32*32
  if (idx >= BB * 2 * 1024) return;
  int b = idx >> 11, o = (idx >> 10) & 1, oy = (idx >> 5) & 31, ox = idx & 31;
  float acc = fb1[o];
  for (int i = 0; i < 2; i++)
    for (int ky = 0; ky < 3; ky++)
      for (int kx = 0; kx < 3; kx++) {
        int y = oy + ky - 1, x = ox + kx - 1;
        if (y >= 0 && y < 32 && x >= 0 && x < 32)
          acc += fused[(size_t)b * 2048 + i * 1024 + y * 32 + x] *
                 fW1[((o * 2 + i) * 3 + ky) * 3 + kx];
      }
  hbuf[idx] = fmaxf(acc, 0.f);
}

__global__ void conv2_kernel(const float* __restrict__ hbuf,
                             const float* __restrict__ fW2, const float* __restrict__ fb2,
                             float* __restrict__ out) {
  int idx = blockIdx.x * blockDim.x + threadIdx.x;   // B*32*32
  if (idx >= BB * 1024) return;
  int b = idx >> 10, oy = (idx >> 5) & 31, ox = idx & 31;
  float acc = fb2[0];
  for (int i = 0; i < 2; i++)
    for (int ky = 0; ky < 3; ky++)
      for (int kx = 0; kx < 3; kx++) {
        int y = oy + ky - 1, x = ox + kx - 1;
        if (y >= 0 && y < 32 && x >= 0 && x < 32)
          acc += hbuf[(size_t)b * 2048 + i * 1024 + y * 32 + x] *
                 fW2[(i * 3 + ky) * 3 + kx];
      }
  out[idx] = acc;
}

extern "C" void kernel_launch(void* const* d_in, const int* in_sizes, int n_in,
                              void* d_out, int out_size, void* d_ws, size_t ws_size,
                              hipStream_t stream) {
  (void)in_sizes; (void)n_in; (void)out_size; (void)ws_size;
  const float* x    = (const float*)d_in[0];
  const float* Wpc  = (const float*)d_in[1];
  const float* bpc  = (const float*)d_in[2];
  const float* Wpf  = (const float*)d_in[3];
  const float* bpf  = (const float*)d_in[4];
  const float* te   = (const float*)d_in[5];
  const float* ln1g = (const float*)d_in[6];
  const float* ln1b = (const float*)d_in[7];
  const float* Wqkv = (const float*)d_in[8];
  const float* Wo   = (const float*)d_in[9];
  const float* bo   = (const float*)d_in[10];
  const float* ln2g = (const float*)d_in[11];
  const float* ln2b = (const float*)d_in[12];
  const float* W1   = (const float*)d_in[13];
  const float* b1   = (const float*)d_in[14];
  const float* W2   = (const float*)d_in[15];
  const float* b2   = (const float*)d_in[16];
  const float* dWc  = (const float*)d_in[17];
  const float* dbc  = (const float*)d_in[18];
  const float* dWf  = (const float*)d_in[19];
  const float* dbf  = (const float*)d_in[20];
  const float* fW1  = (const float*)d_in[21];
  const float* fb1  = (const float*)d_in[22];
  const float* fW2  = (const float*)d_in[23];
  const float* fb2  = (const float*)d_in[24];
  float* out = (float*)d_out;

  // workspace layout (~36 MB of f32)
  float* ws = (float*)d_ws;
  size_t off = 0;
  float* tokens  = ws + off; off += (size_t)BB * NTOK * DIM;   // 1,310,720
  float* ybuf    = ws + off; off += (size_t)BB * NTOK * DIM;   // 1,310,720
  float* qkvbuf  = ws + off; off += (size_t)BB * NTOK * 384;   // 3,932,160 (also MLP hidden)
  float* attnbuf = ws + off; off += (size_t)BB * NTOK * DIM;   // 1,310,720
  float* fineb   = ws + off; off += (size_t)BB * NFT * DIM;    // 1,048,576
  float* cmap    = ws + off; off += (size_t)BB * NCT;
  float* feats   = ws + off; off += (size_t)BB * NFT;
  float* fflat   = ws + off; off += (size_t)BB * NFT;
  float* fused   = ws + off; off += (size_t)BB * 2 * 1024;
  float* hbuf    = ws + off; off += (size_t)BB * 2 * 1024;
  int* maskb = (int*)(ws + off); off += (size_t)BB * NFT;
  int* order = (int*)(ws + off); off += (size_t)BB * NFT;
  int* invb  = (int*)(ws + off); off += (size_t)BB * NFT;
  int* msort = (int*)(ws + off); off += (size_t)BB * NFT;

  // ---- embed + edge mask + stable sort + gather ----
  coarse_embed_kernel<<<(BB * NCT * DIM) / 256, 256, 0, stream>>>(x, Wpc, bpc, te, tokens);
  fine_embed_kernel<<<(BB * NFT * DIM) / 256, 256, 0, stream>>>(x, Wpf, bpf, fineb);
  edge_mask_kernel<<<BB, 1024, 0, stream>>>(x, maskb);
  sort_kernel<<<BB, 1024, 0, stream>>>(maskb, order, invb, msort);
  gather_fine_kernel<<<(BB * NFT * DIM) / 256, 256, 0, stream>>>(fineb, order, msort,
                                                                 te + DIM, tokens);
  // ---- transformer layers ----
  for (int l = 0; l < NLAY; l++) {
    ln_kernel<<<(BB * NTOK) / 8, 256, 0, stream>>>(tokens, ln1g + l * DIM,
                                                   ln1b + l * DIM, ybuf);
    dim3 gq(640, 6);    // N=384 -> 6 strips of 64
    gemm_wmma_kernel<<<gq, 32, 0, stream>>>(ybuf, Wqkv + (size_t)l * DIM * 384,
                                            nullptr, qkvbuf, 384, DIM, 0);
    dim3 ga(NTOK / 16, NHEAD, BB);
    attn_kernel<<<ga, 32, 0, stream>>>(qkvbuf, attnbuf);
    dim3 go(640, 2);    // N=128
    gemm_wmma_kernel<<<go, 32, 0, stream>>>(attnbuf, Wo + (size_t)l * DIM * DIM,
                                            bo + l * DIM, tokens, DIM, DIM, 2);
    ln_kernel<<<(BB * NTOK) / 8, 256, 0, stream>>>(tokens, ln2g + l * DIM,
                                                   ln2b + l * DIM, ybuf);
    float* h1 = qkvbuf;   // reuse (qkv dead here)
    dim3 g1(640, 4);    // N=256
    gemm_wmma_kernel<<<g1, 32, 0, stream>>>(ybuf, W1 + (size_t)l * DIM * 256,
                                            b1 + l * 256, h1, 256, DIM, 1);
    dim3 g2(640, 2);    // N=128
    gemm_wmma_kernel<<<g2, 32, 0, stream>>>(h1, W2 + (size_t)l * 256 * DIM,
                                            b2 + l * DIM, tokens, DIM, 256, 2);
  }
  // ---- decoder ----
  head_kernel<<<(BB * NTOK * 32) / 256, 256, 0, stream>>>(tokens, dWc, dbc, dWf, dbf,
                                                          msort, cmap, feats);
  scatter_kernel<<<(BB * NFT) / 256, 256, 0, stream>>>(feats, invb, fflat);
  fuse_kernel<<<(BB * 1024) / 256, 256, 0, stream>>>(cmap, fflat, fused);
  conv1_kernel<<<(BB * 2 * 1024) / 256, 256, 0, stream>>>(fused, fW1, fb1, hbuf);
  conv2_kernel<<<(BB * 1024) / 256, 256, 0, stream>>>(hbuf, fW2, fb2, out);
}